// CGM_4432406249585
// MI455X (gfx1250) — compile-verified
//
#include <hip/hip_runtime.h>

typedef __attribute__((ext_vector_type(2))) float v2f;
typedef __attribute__((ext_vector_type(8))) float v8f;

#define BATCH 8
#define C_DIM 256
#define K_CLS 19
#define HW_N  16384
#define CT    16          // rows of C per block (one WMMA M-tile)
#define WAVES 8
#define SEG   (HW_N / WAVES)   // 2048 hw positions per wave

// ---------------------------------------------------------------------------
// Kernel 1: att[b,c,k] = sum_hw feature[b,c,hw]*map[b,k,hw]  (WMMA f32 16x16x4)
//           scale[b,c] = sum_k sigmoid(att[b,c,k]) * gamma[k]
// grid: B * C/16 = 128 blocks, 256 threads (8 waves). Each wave reduces a
// 2048-wide HW segment into a 16x32 f32 tile (classes padded 19->32, the pad
// masked to zero multiplicatively so EXEC stays all-ones for WMMA).
// Cross-wave combine is a fixed-order LDS reduction (deterministic).
// ---------------------------------------------------------------------------
__global__ __launch_bounds__(256) void CGM_att_scale_kernel(
    const float* __restrict__ feature,
    const float* __restrict__ map_,
    const float* __restrict__ gamma,
    float* __restrict__ scale)
{
    __shared__ float att_lds[WAVES][CT][32];   // 16 KB

    const int tid  = threadIdx.x;
    const int lane = tid & 31;
    const int wave = tid >> 5;
    const int b    = blockIdx.x >> 4;          // batch
    const int c0   = (blockIdx.x & 15) << 4;   // C-tile base row

    const int row  = lane & 15;                // M row (A) / N column-class (B)
    const int half = lane >> 4;                // 0: k-steps {0,1}; 1: {2,3}

    // A fragment: lane holds feature[b, c0+row, hw + 2*half + {0,1}] (v2f)
    const float* fA = feature
        + (size_t)(b * C_DIM + c0 + row) * HW_N + 2 * half + wave * SEG;
    // B fragment (classes 0..15): map[b, row, hw + 2*half + {0,1}]
    const float* fB0 = map_
        + (size_t)(b * K_CLS + row) * HW_N + 2 * half + wave * SEG;
    // B fragment (classes 16..31): clamp class, zero via mask (no divergence)
    const int   cls1  = 16 + row;
    const float m1    = (cls1 < K_CLS) ? 1.0f : 0.0f;
    const int   cls1c = (cls1 < K_CLS) ? cls1 : (K_CLS - 1);
    const float* fB1 = map_
        + (size_t)(b * K_CLS + cls1c) * HW_N + 2 * half + wave * SEG;

    v8f acc0 = {};   // att tile, classes 0..15
    v8f acc1 = {};   // att tile, classes 16..31 (only 16..18 real)

    #pragma unroll 4
    for (int hw = 0; hw < SEG; hw += 4) {
        v2f a  = *(const v2f*)(fA  + hw);
        v2f b0 = *(const v2f*)(fB0 + hw);
        v2f b1 = *(const v2f*)(fB1 + hw);
        b1.x *= m1;
        b1.y *= m1;
        // D = A(16x4 f32) * B(4x16 f32) + C   -> v_wmma_f32_16x16x4_f32
        acc0 = __builtin_amdgcn_wmma_f32_16x16x4_f32(
            false, a, false, b0, (short)0, acc0, false, false);
        acc1 = __builtin_amdgcn_wmma_f32_16x16x4_f32(
            false, a, false, b1, (short)0, acc1, false, false);
    }

    // D layout: VGPR v, lane L -> M = v + 8*(L>>4), N = L&15
    #pragma unroll
    for (int v = 0; v < 8; ++v) {
        const int m = v + 8 * half;
        att_lds[wave][m][row]      = acc0[v];
        att_lds[wave][m][row + 16] = acc1[v];
    }
    __syncthreads();

    // Deterministic gate: threads 0..15 each finish one C row.
    if (tid < CT) {
        float s = 0.0f;
        for (int k = 0; k < K_CLS; ++k) {
            float a = 0.0f;
            #pragma unroll
            for (int w = 0; w < WAVES; ++w) a += att_lds[w][tid][k];
            const float sig = 1.0f / (1.0f + __expf(-a));
            s += sig * gamma[k];
        }
        scale[b * C_DIM + c0 + tid] = s;
    }
}

// ---------------------------------------------------------------------------
// Kernel 2: out[b,c,:] = feature[b,c,:] * (1 + scale[b,c])
// grid: B*C = 2048 blocks; float4 streaming, fully coalesced (BW-bound pass).
// ---------------------------------------------------------------------------
__global__ __launch_bounds__(256) void CGM_apply_kernel(
    const float* __restrict__ feature,
    const float* __restrict__ scale,
    float* __restrict__ out)
{
    const int    bc   = blockIdx.x;
    const float  s    = 1.0f + scale[bc];
    const size_t base = (size_t)bc * HW_N;
    const float4* __restrict__ src = (const float4*)(feature + base);
    float4* __restrict__       dst = (float4*)(out + base);

    #pragma unroll 4
    for (int i = threadIdx.x; i < HW_N / 4; i += 256) {
        float4 v = src[i];
        v.x *= s; v.y *= s; v.z *= s; v.w *= s;
        dst[i] = v;
    }
}

extern "C" void kernel_launch(void* const* d_in, const int* in_sizes, int n_in,
                              void* d_out, int out_size, void* d_ws, size_t ws_size,
                              hipStream_t stream) {
    const float* feature = (const float*)d_in[0];   // [8,256,128,128] f32
    const float* map_    = (const float*)d_in[1];   // [8,19,128,128]  f32
    const float* gamma   = (const float*)d_in[2];   // [19]            f32
    float*       out     = (float*)d_out;           // [8,256,128,128] f32
    float*       scale   = (float*)d_ws;            // [8*256] f32 scratch

    CGM_att_scale_kernel<<<BATCH * (C_DIM / CT), 256, 0, stream>>>(
        feature, map_, gamma, scale);
    CGM_apply_kernel<<<BATCH * C_DIM, 256, 0, stream>>>(
        feature, scale, out);
}